// QuantumAttentionHead_65481071404559
// MI455X (gfx1250) — compile-verified
//
#include <hip/hip_runtime.h>
#include <hip/hip_bf16.h>

// out[b,i] = cos(token[b,i] + thetas[i])
// Derivation: product circuit of RX gates on |0..0>; RX(a)RX(b)=RX(a+b);
// <Z> after RX(phi)|0> is cos(phi).  Fully f32-exact.
//
// Per 16x16 tile: ang[m,n] = token[m,n] + theta[n] is computed on the matrix
// pipe via 3 chained V_WMMA_F32_16X16X4_F32 ops with diagonal-identity B
// blocks (the WMMA performs the cross-lane transpose from A layout to C/D
// layout), then v_cos_f32 on the 8 accumulator VGPRs, then store.
//
// Columns 10..15 of the tile are never stored, so A entries feeding those
// columns may hold arbitrary finite values -> loads are unconditional with a
// clamped (aligned, in-bounds) column index: zero divergent control flow
// before the WMMAs (EXEC stays all-1s as required).

typedef __attribute__((ext_vector_type(2))) float v2f;
typedef __attribute__((ext_vector_type(8))) float v8f;

#define N_WIRES 10
#define WAVES_PER_BLOCK 8

__global__ __launch_bounds__(WAVES_PER_BLOCK * 32)
void qah_cos_wmma_kernel(const float* __restrict__ token,
                         const float* __restrict__ thetas,
                         float* __restrict__ out,
                         int ntiles) {
    const int lane = threadIdx.x & 31;
    const int wave = threadIdx.x >> 5;
    const int tile = blockIdx.x * WAVES_PER_BLOCK + wave;
    if (tile >= ntiles) return;               // uniform per-wave exit

    const int rowBase = tile * 16;
    const int half = lane >> 4;               // 0: lanes 0-15, 1: lanes 16-31
    const int r    = lane & 15;               // A row (M) for this lane; N column for B/D
    const int row  = rowBase + r;             // batch row this lane supplies for A

    const float* rowp = token + row * N_WIRES;

    // Accumulator (C matrix) starts at zero.
    v8f acc = {0.f, 0.f, 0.f, 0.f, 0.f, 0.f, 0.f, 0.f};

    // 3 chained K=4 f32 WMMAs: D[m,n] = sum_g sum_k A_g[m,k] * B_g[k,n]
    //   A_g[m,k] = token[m, 4g+k] + theta[4g+k]
    //   B_g[k,n] = 1 iff n == 4g+k   (diagonal identity block)
    // => D[m,n] = token[m,n] + theta[n] for n < 10.
#pragma unroll
    for (int g = 0; g < 3; ++g) {
        // This lane holds A VGPR0/VGPR1 = A_g[r][2*half + 0/1]  (ISA 16x4 f32 A layout)
        const int c0 = 4 * g + 2 * half;          // always even
        const int cc = (c0 < N_WIRES) ? c0 : 8;   // clamp: aligned + in-bounds; garbage OK

        // One b64 load each: token pair + theta pair (8-byte aligned).
        v2f t  = *(const v2f*)(rowp + cc);
        v2f th = *(const v2f*)(thetas + cc);
        v2f a  = t + th;

        // B layout mirrors A's half-lane split: VGPR0 row k = c0, VGPR1 row
        // k = c0+1, column n = lane&15.
        v2f b;
        b.x = (r == c0)     ? 1.0f : 0.0f;
        b.y = (r == c0 + 1) ? 1.0f : 0.0f;

        acc = __builtin_amdgcn_wmma_f32_16x16x4_f32(
            /*neg_a=*/false, a, /*neg_b=*/false, b,
            /*c_mod=*/(short)0, acc, /*reuse_a=*/false, /*reuse_b=*/false);
    }

    // C/D layout: VGPR v holds D[M = v + 8*half][N = r].
    if (r < N_WIRES) {
        constexpr float INV_2PI = 0.15915494309189535f;  // v_cos_f32 unit = revolutions
        float* colp = out + (rowBase + 8 * half) * N_WIRES + r;
#pragma unroll
        for (int v = 0; v < 8; ++v) {
            colp[v * N_WIRES] = __builtin_amdgcn_cosf(acc[v] * INV_2PI);
        }
    }
}

extern "C" void kernel_launch(void* const* d_in, const int* in_sizes, int n_in,
                              void* d_out, int out_size, void* d_ws, size_t ws_size,
                              hipStream_t stream) {
    (void)n_in; (void)d_ws; (void)ws_size; (void)out_size;
    const float* token  = (const float*)d_in[0];   // (BATCH, 10) f32
    const float* thetas = (const float*)d_in[1];   // (10,) f32
    float* out = (float*)d_out;                    // (BATCH, 10) f32

    const int batch  = in_sizes[0] / N_WIRES;      // 16384 (multiple of 16)
    const int ntiles = batch >> 4;                 // 1024 full tiles
    const int blocks = (ntiles + WAVES_PER_BLOCK - 1) / WAVES_PER_BLOCK;

    qah_cos_wmma_kernel<<<blocks, WAVES_PER_BLOCK * 32, 0, stream>>>(
        token, thetas, out, ntiles);
}